// res_gcn_d_79714593014269
// MI455X (gfx1250) — compile-verified
//
#include <hip/hip_runtime.h>
#include <hip/hip_bf16.h>

typedef __attribute__((ext_vector_type(2))) float v2f;
typedef __attribute__((ext_vector_type(8))) float v8f;
typedef __attribute__((ext_vector_type(4))) int   v4i;

#define BB   8
#define NN   4096
#define CC   128
#define KK   20
#define LPITCH 129   // LDS pitch for transposed weights (avoids 64-bank conflicts)

#if defined(__has_builtin)
#if __has_builtin(__builtin_amdgcn_global_load_async_to_lds_b128)
#define USE_ASYNC_LDS 1
#endif
#endif
#ifndef USE_ASYNC_LDS
#define USE_ASYNC_LDS 0
#endif

#define AS1 __attribute__((address_space(1)))
#define AS3 __attribute__((address_space(3)))

__device__ __forceinline__ void wait_async_zero() {
#if USE_ASYNC_LDS
#if __has_builtin(__builtin_amdgcn_s_wait_asynccnt)
    __builtin_amdgcn_s_wait_asynccnt(0);
#else
    asm volatile("s_wait_asynccnt 0x0" ::: "memory");
#endif
#endif
}

__device__ __forceinline__ float leaky01(float x) { return fmaxf(x, 0.01f * x); }

// ---------------------------------------------------------------------------
// Tiled transpose: in[b][r][s] -> out[b][s][r]
// ---------------------------------------------------------------------------
__global__ __launch_bounds__(256) void transpose_kernel(
    const float* __restrict__ in, float* __restrict__ out, int R, int S)
{
    __shared__ float tile[32][33];
    const int b  = blockIdx.z;
    const int r0 = blockIdx.y * 32;
    const int s0 = blockIdx.x * 32;
    const int tx = threadIdx.x & 31;
    const int ty = threadIdx.x >> 5;   // 0..7
    const float* ib = in + (size_t)b * R * S;
    float* ob = out + (size_t)b * R * S;
#pragma unroll
    for (int i = 0; i < 32; i += 8)
        tile[ty + i][tx] = ib[(size_t)(r0 + ty + i) * S + s0 + tx];
    __syncthreads();
#pragma unroll
    for (int i = 0; i < 32; i += 8)
        ob[(size_t)(s0 + ty + i) * R + r0 + tx] = tile[tx][ty + i];
}

// ---------------------------------------------------------------------------
// xyz [B,3,N] -> float4 (x,y,z,|p|^2) [B,N]
// ---------------------------------------------------------------------------
__global__ __launch_bounds__(256) void xyz_prep_kernel(
    const float* __restrict__ xyz, float4* __restrict__ xyzq)
{
    const int b = blockIdx.y;
    const int n = blockIdx.x * 256 + threadIdx.x;
    const float x = xyz[((size_t)b * 3 + 0) * NN + n];
    const float y = xyz[((size_t)b * 3 + 1) * NN + n];
    const float z = xyz[((size_t)b * 3 + 2) * NN + n];
    float4 o; o.x = x; o.y = y; o.z = z; o.w = x * x + y * y + z * z;
    xyzq[(size_t)b * NN + n] = o;
}

// ---------------------------------------------------------------------------
// KNN, faithful to the reference: k+1 LARGEST squared distances, drop slot 0.
// Ranking key r(j) = |p_j|^2 - 2 q.p_j (adding |q|^2 doesn't change order).
// One wave per query row; batch xyz staged into LDS with
// GLOBAL_LOAD_ASYNC_TO_LDS_B128 (ASYNCcnt) when available.
// Per-lane top-21 sorted register list (fully unrolled -> stays in VGPRs),
// then 21 rounds of wave32 argmax merge via shuffles.
// ---------------------------------------------------------------------------
__global__ __launch_bounds__(256) void knn_kernel(
    const float4* __restrict__ xyzq, int* __restrict__ idxOut)
{
    extern __shared__ float4 spts[];
    const int b = blockIdx.y;
    const int rowBlk = blockIdx.x;
    const float4* src = xyzq + (size_t)b * NN;
#if USE_ASYNC_LDS
    for (int t = threadIdx.x; t < NN; t += 256)
        __builtin_amdgcn_global_load_async_to_lds_b128(
            (AS1 v4i*)(src + t), (AS3 v4i*)(spts + t), 0, 0);
    wait_async_zero();
#else
    for (int t = threadIdx.x; t < NN; t += 256) spts[t] = src[t];
#endif
    __syncthreads();

    const int wave = threadIdx.x >> 5;
    const int lane = threadIdx.x & 31;

    for (int rr = 0; rr < 8; ++rr) {
        const int row = rowBlk * 64 + wave * 8 + rr;
        const float4 q = spts[row];

        float tv[21]; int ti[21];          // sorted ascending, min at [0]
#pragma unroll
        for (int s = 0; s < 21; ++s) { tv[s] = -1e30f; ti[s] = -1; }

        for (int j0 = 0; j0 < NN; j0 += 32) {
            const int j = j0 + lane;       // lane-disjoint -> unique indices
            const float4 p = spts[j];
            const float r = p.w - 2.0f * (q.x * p.x + q.y * p.y + q.z * p.z);
            if (r > tv[0]) {
                float cv = r; int ci = j;
                bool placed = false;
#pragma unroll
                for (int s = 0; s < 20; ++s) {
                    const bool shift = !placed && (cv > tv[s + 1]);
                    if (shift) { tv[s] = tv[s + 1]; ti[s] = ti[s + 1]; }
                    else if (!placed) { tv[s] = cv; ti[s] = ci; placed = true; }
                }
                if (!placed) { tv[20] = cv; ti[20] = ci; }
            }
        }

        // Merge: rounds 1..20 emit indices (round 0 = dropped farthest).
        int* orow = idxOut + ((size_t)b * NN + row) * KK;
        for (int r = 0; r < KK + 1; ++r) {
            float mv = tv[20]; int mi = ti[20];
#pragma unroll
            for (int off = 16; off >= 1; off >>= 1) {
                const float ov = __shfl_xor(mv, off, 32);
                const int   oi = __shfl_xor(mi, off, 32);
                if (ov > mv || (ov == mv && oi < mi)) { mv = ov; mi = oi; }
            }
            const bool winner = (tv[20] == mv) && (ti[20] == mi);
            if (winner) {
#pragma unroll
                for (int s = 20; s >= 1; --s) { tv[s] = tv[s - 1]; ti[s] = ti[s - 1]; }
                tv[0] = -1e30f; ti[0] = -1;
            }
            if (r >= 1 && lane == 0) orow[r - 1] = mi;
        }
    }
}

// ---------------------------------------------------------------------------
// sT[b,n,:] = sum_j leaky(pts[b, idx[b,n,j], :])   (wave per point)
// ---------------------------------------------------------------------------
__global__ __launch_bounds__(256) void gather_sum_kernel(
    const float* __restrict__ pts, const int* __restrict__ idx,
    float* __restrict__ sT)
{
    const int wave = threadIdx.x >> 5;
    const int lane = threadIdx.x & 31;
    const size_t point = (size_t)blockIdx.x * 8 + wave;
    const size_t base = (point >> 12) << 12;     // b * NN  (NN = 4096)
    const int* ip = idx + point * KK;
    float4 acc = {0.f, 0.f, 0.f, 0.f};
#pragma unroll 4
    for (int j = 0; j < KK; ++j) {
        const int t = ip[j];                     // wave-uniform -> scalar load
        const float4 v =
            reinterpret_cast<const float4*>(pts + (base + (size_t)t) * CC)[lane];
        acc.x += leaky01(v.x); acc.y += leaky01(v.y);
        acc.z += leaky01(v.z); acc.w += leaky01(v.w);
    }
    reinterpret_cast<float4*>(sT + point * CC)[lane] = acc;
}

// ---------------------------------------------------------------------------
// One K=128 pass of the fused GEMM: acc += leaky?(A[row,:]) x Wt.
// Software-pipelined: the next A fragment is loaded before the current one is
// consumed, so s_wait_loadcnt lands behind 8 WMMAs + the B ds-loads.
// ---------------------------------------------------------------------------
template <bool LEAKY>
__device__ __forceinline__ void gemm_pass(const float* __restrict__ ap,
                                          const float* __restrict__ wt,
                                          int khalf, int mrow, v8f acc[8])
{
    auto step = [&](float2 av, int k0) {
        v2f a;
        a.x = LEAKY ? leaky01(av.x) : av.x;
        a.y = LEAKY ? leaky01(av.y) : av.y;
        const float* wb = wt + (k0 + khalf * 2) * LPITCH + mrow;
#pragma unroll
        for (int t = 0; t < 8; ++t) {
            v2f bf; bf.x = wb[t * 16]; bf.y = wb[t * 16 + LPITCH];
            acc[t] = __builtin_amdgcn_wmma_f32_16x16x4_f32(
                false, a, false, bf, (short)0, acc[t], false, false);
        }
    };
    float2 av = *reinterpret_cast<const float2*>(ap);
#pragma unroll 4
    for (int k0 = 0; k0 < CC - 4; k0 += 4) {
        const float2 nx = *reinterpret_cast<const float2*>(ap + k0 + 4);
        step(av, k0);
        av = nx;
    }
    step(av, CC - 4);
}

// ---------------------------------------------------------------------------
// Fused residual GEMM using V_WMMA_F32_16X16X4_F32.
//   out[r,:] = scale * ( leaky(A1[r,:])*W1^T + maybeLeaky(A2[r,:])*W2^T
//                        + b1 + c2*b2 ) + A1[r,:]
// 8 waves/WG, each wave owns 16 rows x 128 outputs (8 accumulator tiles).
// Both weight matrices live transposed in LDS (Wt[c][o], pitch 129).
// ---------------------------------------------------------------------------
__global__ __launch_bounds__(256) void fused_gemm_kernel(
    const float* __restrict__ A1, const float* __restrict__ A2, int leakyA2,
    const float* __restrict__ W1, const float* __restrict__ W2,
    const float* __restrict__ b1, const float* __restrict__ b2,
    float c2, float scale, float* __restrict__ out)
{
    extern __shared__ float lds[];
    float* Wt1 = lds;                 // [CC][LPITCH]
    float* Wt2 = lds + CC * LPITCH;

    for (int t = threadIdx.x; t < CC * CC; t += 256) {
        const int o = t >> 7, c = t & (CC - 1);
        Wt1[c * LPITCH + o] = W1[t];
        Wt2[c * LPITCH + o] = W2[t];
    }
    __syncthreads();

    const int wave  = threadIdx.x >> 5;
    const int lane  = threadIdx.x & 31;
    const int mrow  = lane & 15;      // M (and N) index within 16-wide tile
    const int khalf = lane >> 4;      // which K pair this lane carries
    const size_t rowBase = (size_t)blockIdx.x * 128 + (size_t)wave * 16;

    v8f acc[8];
    const v8f zero = {0.f, 0.f, 0.f, 0.f, 0.f, 0.f, 0.f, 0.f};
#pragma unroll
    for (int t = 0; t < 8; ++t) acc[t] = zero;

    const float* a1p = A1 + (rowBase + mrow) * CC + khalf * 2;
    const float* a2p = A2 + (rowBase + mrow) * CC + khalf * 2;
    __builtin_prefetch(a1p, 0, 3);
    __builtin_prefetch(a2p, 0, 3);

    // Pass 1: leaky(A1) x W1^T
    gemm_pass<true>(a1p, Wt1, khalf, mrow, acc);
    // Pass 2: A2 x W2^T (leaky for blocks 1-2; neighbor-sum already activated)
    if (leakyA2) gemm_pass<true>(a2p, Wt2, khalf, mrow, acc);
    else         gemm_pass<false>(a2p, Wt2, khalf, mrow, acc);

    // Epilogue: D layout -> VGPR v: lanes 0-15 = M=v, lanes 16-31 = M=v+8
#pragma unroll
    for (int t = 0; t < 8; ++t) {
        const int n = t * 16 + mrow;
        const float bo = b1[n] + c2 * b2[n];
#pragma unroll
        for (int v = 0; v < 8; ++v) {
            const size_t r = rowBase + (size_t)(v + khalf * 8);
            const float sc = A1[r * CC + n];          // residual (pre-activation)
            out[r * CC + n] = scale * (acc[t][v] + bo) + sc;
        }
    }
}

// ---------------------------------------------------------------------------
extern "C" void kernel_launch(void* const* d_in, const int* in_sizes, int n_in,
                              void* d_out, int out_size, void* d_ws, size_t ws_size,
                              hipStream_t stream)
{
    (void)in_sizes; (void)n_in; (void)out_size; (void)ws_size;
    const float* xyz    = (const float*)d_in[0];
    const float* points = (const float*)d_in[1];
    const float* W1     = (const float*)d_in[2];
    const float* b1     = (const float*)d_in[3];
    const float* W2     = (const float*)d_in[4];
    const float* b2     = (const float*)d_in[5];
    // d_in[6] = k (always 20 per setup)

    char* ws = (char*)d_ws;
    const size_t BNC = (size_t)BB * NN * CC;                 // 4.19M floats
    float*  ptsA   = (float*)ws;                             // 16 MB
    float*  ptsB   = (float*)(ws + BNC * 4);                 // 16 MB
    float*  sT     = (float*)(ws + 2 * BNC * 4);             // 16 MB
    float4* xyzq   = (float4*)(ws + 3 * BNC * 4);            // 512 KB
    int*    idxbuf = (int*)(ws + 3 * BNC * 4 + (size_t)BB * NN * 16); // 2.6 MB

    // 1) points [B,C,N] -> ptsA [B,N,C]
    transpose_kernel<<<dim3(NN / 32, CC / 32, BB), 256, 0, stream>>>(points, ptsA, CC, NN);
    // 2) xyz -> (x,y,z,|p|^2)
    xyz_prep_kernel<<<dim3(NN / 256, BB), 256, 0, stream>>>(xyz, xyzq);
    // 3) top-k (largest) neighbor indices
    knn_kernel<<<dim3(NN / 64, BB), 256, NN * sizeof(float4), stream>>>(xyzq, idxbuf);
    // 4) neighbor gather-sum with leaky (linearizes the k-neighbor einsum)
    gather_sum_kernel<<<(BB * NN) / 8, 256, 0, stream>>>(ptsA, idxbuf, sT);

    const size_t gemmLds = (size_t)2 * CC * LPITCH * sizeof(float); // 132 KB
    // 5) block 0: (W1*leaky(p) + W2*s + b1 + 20*b2)/21 + p
    fused_gemm_kernel<<<(BB * NN) / 128, 256, gemmLds, stream>>>(
        ptsA, sT, 0, W1, W2, b1, b2, 20.f, 1.f / 21.f, ptsB);
    // 6) block 1: 0.5*((W1+W2)*leaky(p) + b1 + b2) + p
    fused_gemm_kernel<<<(BB * NN) / 128, 256, gemmLds, stream>>>(
        ptsB, ptsB, 1, W1 + CC * CC, W2 + CC * CC, b1 + CC, b2 + CC, 1.f, 0.5f, ptsA);
    // 7) block 2
    fused_gemm_kernel<<<(BB * NN) / 128, 256, gemmLds, stream>>>(
        ptsA, ptsA, 1, W1 + 2 * CC * CC, W2 + 2 * CC * CC, b1 + 2 * CC, b2 + 2 * CC,
        1.f, 0.5f, ptsB);
    // 8) ptsB [B,N,C] -> d_out [B,C,N]
    transpose_kernel<<<dim3(CC / 32, NN / 32, BB), 256, 0, stream>>>(
        ptsB, (float*)d_out, NN, CC);
}